// JointSparseEmbedding_3496103379186
// MI455X (gfx1250) — compile-verified
//
#include <hip/hip_runtime.h>

// JointSparseEmbedding: out[b,t,0:128] = fp16(weights[idx[b,t] + offsets[t], 0:128])
//
// Bandwidth-bound gather+cast. MI455X roofline: ~436 MB gathered reads +
// ~218 MB NT writes over 23.3 TB/s => ~28 us floor, zero FLOPs (WMMA N/A).
//
// v2 changes (driven by round-1 disasm): the per-lookup scalar overhead
// (%26 magic multiply + 64-bit vector address math, ~30 VALU/SALU per
// 768 bytes moved) was the same order as the memory floor. Now each wave
// handles 8 lookups:
//   * lanes 0..7 do the index load, %26, and offsets[] gather in parallel
//     (one vector op per 8 lookups instead of 8 serial scalar sequences)
//   * v_readlane broadcasts each row into an SGPR -> SALU base + b128 load
//     with a single shared lane*16 voffset
//   * 8 independent global_load_b128 in flight per wave before first use
//   * 8 stores share one address, differing only in the 24-bit immediate
//     offset (i*256 B); stores are non-temporal to protect L2 for row reuse
//
// wave32 == one 512B row per wave per iteration (32 lanes x b128 = 512 B).

typedef float    vfloat4 __attribute__((ext_vector_type(4)));
typedef _Float16 vhalf4  __attribute__((ext_vector_type(4)));

#define JSE_NUM_TABLES 26
#define JSE_EMBED_DIM  128
#define JSE_V4_PER_ROW (JSE_EMBED_DIM / 4)   // 32 float4 / half4 per row
#define JSE_UNROLL     8                     // lookups per wave

__global__ __launch_bounds__(256) void JointSparseEmbedding_gather_kernel(
    const int*   __restrict__ cat,      // [B*26] categorical indices (flat)
    const float* __restrict__ weights,  // [26*100000, 128] fp32 joint table
    const int*   __restrict__ offsets,  // [26] table base offsets
    _Float16*    __restrict__ out,      // [B*26, 128] fp16
    int num_lookups)                    // B * 26
{
    const int lane = (int)(threadIdx.x & 31u);
    const int wave = (int)((blockIdx.x * blockDim.x + threadIdx.x) >> 5);
    const int base = wave * JSE_UNROLL;
    if (base >= num_lookups) return;

    // Lane-parallel index fetch for this wave's 8 lookups.
    // Lanes >= 8 clamp to lane 7 so the loads stay in bounds; their values
    // are never consumed (readlane only reads lanes 0..7).
    const int li   = (lane < JSE_UNROLL) ? lane : (JSE_UNROLL - 1);
    const int look = base + li;
    const int t    = look % JSE_NUM_TABLES;       // one vector modulo / 8 lookups
    const int row  = cat[look] + offsets[t];      // per-lane row ids

    const vfloat4* wv = (const vfloat4*)weights;

    if (base + JSE_UNROLL <= num_lookups) {
        // ---- fast path: full group of 8 ----
        vfloat4 v[JSE_UNROLL];
#pragma unroll
        for (int i = 0; i < JSE_UNROLL; ++i) {
            // Broadcast row i into an SGPR -> scalar base + lane*16 voffset.
            const int r = __builtin_amdgcn_readlane(row, i);
            v[i] = wv[(size_t)r * JSE_V4_PER_ROW + lane];   // global_load_b128
        }

        vhalf4* dst = (vhalf4*)out + (size_t)base * JSE_V4_PER_ROW + lane;
#pragma unroll
        for (int i = 0; i < JSE_UNROLL; ++i) {
            const vhalf4 h = __builtin_convertvector(v[i], vhalf4); // RNE cvt
            // Same base address for all 8 stores; i*256 B folds into the
            // 24-bit instruction immediate. NT store (write-once stream).
            __builtin_nontemporal_store(h, dst + i * JSE_V4_PER_ROW);
        }
    } else {
        // ---- tail: per-lookup fallback ----
        for (int k = base; k < num_lookups; ++k) {
            const int r = __builtin_amdgcn_readfirstlane(
                cat[k] + offsets[k % JSE_NUM_TABLES]);
            const vfloat4 v = wv[(size_t)r * JSE_V4_PER_ROW + lane];
            const vhalf4  h = __builtin_convertvector(v, vhalf4);
            __builtin_nontemporal_store(
                h, (vhalf4*)out + (size_t)k * JSE_V4_PER_ROW + lane);
        }
    }
}

extern "C" void kernel_launch(void* const* d_in, const int* in_sizes, int n_in,
                              void* d_out, int out_size, void* d_ws, size_t ws_size,
                              hipStream_t stream) {
    const int*   cat     = (const int*)d_in[0];   // [B, 26]
    const float* weights = (const float*)d_in[1]; // [2.6M, 128]
    const int*   offsets = (const int*)d_in[2];   // [26]
    _Float16*    out     = (_Float16*)d_out;      // [B, 26, 128] fp16

    const int num_lookups = in_sizes[0];          // B * 26 = 851,968

    // 8 waves per 256-thread block, 8 lookups per wave -> 64 lookups/block.
    const int threads         = 256;
    const int lookupsPerBlock = (threads / 32) * JSE_UNROLL;
    const int blocks = (num_lookups + lookupsPerBlock - 1) / lookupsPerBlock;

    JointSparseEmbedding_gather_kernel<<<blocks, threads, 0, stream>>>(
        cat, weights, offsets, out, num_lookups);
}